// SimpleGNN_78718160601833
// MI455X (gfx1250) — compile-verified
//
#include <hip/hip_runtime.h>

#define C_DIM 128
#define NUM_G 512
#define BN_EPS 1e-5f

typedef float v2f __attribute__((ext_vector_type(2)));
typedef float v8f __attribute__((ext_vector_type(8)));
typedef unsigned int u32x4 __attribute__((ext_vector_type(4)));
typedef int i32x4 __attribute__((ext_vector_type(4)));
typedef int i32x8 __attribute__((ext_vector_type(8)));

// ---------------- utility ----------------
__global__ void k_zero(float* __restrict__ p, int n) {
  int i = blockIdx.x * blockDim.x + threadIdx.x;
  if (i < n) p[i] = 0.0f;
}

__global__ void k_copy4(const float4* __restrict__ src, float4* __restrict__ dst, int n4) {
  int i = blockIdx.x * blockDim.x + threadIdx.x;
  if (i < n4) dst[i] = src[i];
}

// ---------------- encoder: h0 = x@node_W + node_b + pe@pe_W + pe_b ----------------
__global__ void k_init_h(const float* __restrict__ x, const float* __restrict__ pe,
                         const float* __restrict__ nW, const float* __restrict__ nb,
                         const float* __restrict__ pW, const float* __restrict__ pb,
                         float* __restrict__ h, int N) {
  int idx = blockIdx.x * blockDim.x + threadIdx.x;
  if (idx >= N * C_DIM) return;
  int n = idx >> 7, c = idx & (C_DIM - 1);
  float acc = x[n] * nW[c] + nb[c] + pb[c];
#pragma unroll
  for (int k = 0; k < 8; ++k) acc += pe[n * 8 + k] * pW[k * C_DIM + c];
  h[idx] = acc;
}

// ---------------- degree / dinv ----------------
__global__ void k_degree(const int* __restrict__ ei, float* __restrict__ deg, int E) {
  int e = blockIdx.x * blockDim.x + threadIdx.x;
  if (e < E) unsafeAtomicAdd(&deg[ei[e]], 1.0f);
}

__global__ void k_dinv(float* __restrict__ deg, int N) {
  int i = blockIdx.x * blockDim.x + threadIdx.x;
  if (i < N) {
    float d = deg[i];
    deg[i] = (d > 0.0f) ? rsqrtf(d) : 0.0f;
  }
}

// ---------------- edge scatter: tmp[row] += h[col] * dinv[row]*dinv[col] ----------------
// One wave (32 lanes) per edge; each lane owns 4 channels (float4).
__global__ void k_edge(const int* __restrict__ ei, const float* __restrict__ dinv,
                       const float* __restrict__ h, float* __restrict__ tmp, int E) {
  int gid = blockIdx.x * blockDim.x + threadIdx.x;
  int e = gid >> 5, lane = gid & 31;
  if (e >= E) return;
  int r = ei[e];
  int c = ei[E + e];
  float norm = dinv[r] * dinv[c];
  float4 v = ((const float4*)(h + (size_t)c * C_DIM))[lane];
  float* dst = tmp + (size_t)r * C_DIM + lane * 4;
  unsafeAtomicAdd(dst + 0, v.x * norm);
  unsafeAtomicAdd(dst + 1, v.y * norm);
  unsafeAtomicAdd(dst + 2, v.z * norm);
  unsafeAtomicAdd(dst + 3, v.w * norm);
}

// ---------------- GEMM (in-place): tmp = tmp @ W + b, with fused BN partial sums ----------
// Block = 256 threads = 8 waves, 16 rows/wave -> 128 rows/block.
// Wave computes 16x128 via 8 accumulator tiles of V_WMMA_F32_16X16X4_F32, K=128 in 32 steps.
// Weight tile (128x128 f32 = 64KB) staged into LDS by the Tensor Data Mover.
__global__ __launch_bounds__(256) void k_gemm_bn(const float* __restrict__ A,
                                                 const float* __restrict__ Wg,
                                                 const float* __restrict__ bias,
                                                 float* __restrict__ out,
                                                 float* __restrict__ bnsum,
                                                 float* __restrict__ bnsq, int N) {
  __shared__ float Wlds[C_DIM * C_DIM];  // 64 KB: weight matrix, row-major [k][n]

#if __has_builtin(__builtin_amdgcn_tensor_load_to_lds)
  // TDM stage: one descriptor per block, issued by wave 0; D# per CDNA5 ISA.
  if (threadIdx.x == 0) {
    unsigned long long ga = (unsigned long long)(const void*)Wg;
    unsigned ldsOff = (unsigned)(size_t)(void*)Wlds;  // low 32 bits = LDS byte offset
    u32x4 g0;
    g0[0] = 1u;                                   // count=1, user descriptor
    g0[1] = ldsOff;                               // lds_addr (bytes)
    g0[2] = (unsigned)ga;                         // global_addr[31:0]
    g0[3] = ((unsigned)(ga >> 32) & 0x01FFFFFFu)  // global_addr[56:32]
            | 0x80000000u;                        // type=2 ("image") in bits 127:126
    i32x8 g1;
    g1[0] = 0x20000;      // workgroup_mask=0, data_size=2 (4B), no barrier/iter/pad
    g1[1] = 0x40000000;   // tensor_dim0[15:0]=16384 in bits 63:48 (=0x4000<<16)
    g1[2] = 0x00010000;   // tensor_dim0 hi=0, tensor_dim1[15:0]=1
    g1[3] = 0x40000000;   // tensor_dim1 hi=0, tile_dim0=16384
    g1[4] = 1;            // tile_dim1=1, tile_dim2=0
    g1[5] = 16384;        // tensor_dim0_stride low 32
    g1[6] = 0;            // stride hi, tensor_dim1_stride low
    g1[7] = 0;
    i32x4 gz = {0, 0, 0, 0};
#if defined(__clang_major__) && (__clang_major__ >= 23)
    i32x8 gz8 = {0, 0, 0, 0, 0, 0, 0, 0};
    __builtin_amdgcn_tensor_load_to_lds(g0, g1, gz, gz, gz8, 0);
#else
    __builtin_amdgcn_tensor_load_to_lds(g0, g1, gz, gz, 0);
#endif
    __builtin_amdgcn_s_wait_tensorcnt(0);
  }
#else
  for (int j = threadIdx.x; j < (C_DIM * C_DIM) / 4; j += blockDim.x)
    ((float4*)Wlds)[j] = ((const float4*)Wg)[j];
#endif
  __syncthreads();

  const int lane = threadIdx.x & 31;
  const int wave = threadIdx.x >> 5;
  const int row0 = (blockIdx.x * 8 + wave) * 16;
  if (row0 >= N) return;

  const int m = lane & 15;           // row within 16-row tile (A) / col within 16 (B,D)
  const int khalf = (lane >> 4) * 2; // K offset 0 or 2 per ISA A/B layout

  v8f acc[8];
#pragma unroll
  for (int t = 0; t < 8; ++t) acc[t] = {};

  // Loop-invariant tail guard: clamp the row pointer, zero via mask multiply.
  const int rowA = row0 + m;
  const float aMask = (rowA < N) ? 1.0f : 0.0f;
  const int rowC = (rowA < N) ? rowA : (N - 1);
  const float* aptr = A + (size_t)rowC * C_DIM + khalf;

  // Prefetch this lane's A row (4 cachelines) ahead of the K loop.
#pragma unroll
  for (int pk = 0; pk < C_DIM; pk += 32) __builtin_prefetch(aptr + pk, 0, 3);

  for (int k0 = 0; k0 < C_DIM; k0 += 4) {
    v2f a = *(const v2f*)(aptr + k0);
    a.x *= aMask;
    a.y *= aMask;
#pragma unroll
    for (int t = 0; t < 8; ++t) {
      v2f b;
      b.x = Wlds[(k0 + khalf) * C_DIM + t * 16 + m];
      b.y = Wlds[(k0 + khalf + 1) * C_DIM + t * 16 + m];
      acc[t] = __builtin_amdgcn_wmma_f32_16x16x4_f32(false, a, false, b, (short)0,
                                                     acc[t], false, false);
    }
  }

  // Epilogue: add bias, store, accumulate BN partial sums (sum, sum of squares).
  const int rbase = row0 + ((lane >> 4) << 3);  // D layout: lanes16-31 hold rows M=8..15
#pragma unroll
  for (int t = 0; t < 8; ++t) {
    const int col = t * 16 + m;
    const float bb = bias[col];
    float s = 0.0f, sq = 0.0f;
#pragma unroll
    for (int i = 0; i < 8; ++i) {
      int r = rbase + i;
      if (r < N) {
        float v = acc[t][i] + bb;
        out[(size_t)r * C_DIM + col] = v;
        s += v;
        sq += v * v;
      }
    }
    unsafeAtomicAdd(&bnsum[col], s);
    unsafeAtomicAdd(&bnsq[col], sq);
  }
}

// ---------------- BN + ReLU + residual: h += relu((pre-mu)*rsqrt(var+eps)*g + b) ---------
__global__ void k_bn_apply(const float* __restrict__ pre, float* __restrict__ h,
                           const float* __restrict__ bnsum, const float* __restrict__ bnsq,
                           const float* __restrict__ gamma, const float* __restrict__ beta,
                           float invN, int total) {
  int idx = blockIdx.x * blockDim.x + threadIdx.x;
  if (idx >= total) return;
  int c = idx & (C_DIM - 1);
  float mu = bnsum[c] * invN;
  float var = bnsq[c] * invN - mu * mu;
  float y = (pre[idx] - mu) * rsqrtf(var + BN_EPS) * gamma[c] + beta[c];
  h[idx] += fmaxf(y, 0.0f);
}

// ---------------- mean pooling (atomics) ----------------
__global__ void k_pool(const float* __restrict__ h, const int* __restrict__ batch,
                       float* __restrict__ pooled, float* __restrict__ cnt, int N) {
  int gid = blockIdx.x * blockDim.x + threadIdx.x;
  int node = gid >> 5, lane = gid & 31;
  if (node >= N) return;
  int g = batch[node];
  float4 v = ((const float4*)(h + (size_t)node * C_DIM))[lane];
  float* dst = pooled + (size_t)g * C_DIM + lane * 4;
  unsafeAtomicAdd(dst + 0, v.x);
  unsafeAtomicAdd(dst + 1, v.y);
  unsafeAtomicAdd(dst + 2, v.z);
  unsafeAtomicAdd(dst + 3, v.w);
  if (lane == 0) unsafeAtomicAdd(&cnt[g], 1.0f);
}

// ---------------- head: out[g] = relu(mean @ W1 + b1) @ W2 + b2 ----------------
__global__ __launch_bounds__(128) void k_head(const float* __restrict__ pooled,
                                              const float* __restrict__ cnt,
                                              const float* __restrict__ W1,
                                              const float* __restrict__ b1,
                                              const float* __restrict__ W2,
                                              const float* __restrict__ b2,
                                              float* __restrict__ out) {
  int g = blockIdx.x;
  int tid = threadIdx.x;
  __shared__ float pm[C_DIM];
  __shared__ float red[C_DIM];
  float cc = cnt[g];
  cc = (cc < 1.0f) ? 1.0f : cc;
  pm[tid] = pooled[(size_t)g * C_DIM + tid] / cc;
  __syncthreads();
  float acc = b1[tid];
#pragma unroll 8
  for (int k = 0; k < C_DIM; ++k) acc += pm[k] * W1[k * C_DIM + tid];
  acc = fmaxf(acc, 0.0f);
  red[tid] = acc * W2[tid];
  __syncthreads();
  for (int s = 64; s > 0; s >>= 1) {
    if (tid < s) red[tid] += red[tid + s];
    __syncthreads();
  }
  if (tid == 0) out[g] = red[0] + b2[0];
}

// ---------------- host launcher ----------------
extern "C" void kernel_launch(void* const* d_in, const int* in_sizes, int n_in,
                              void* d_out, int out_size, void* d_ws, size_t ws_size,
                              hipStream_t stream) {
  const float* x      = (const float*)d_in[0];
  const float* pe     = (const float*)d_in[1];
  const int*   ei     = (const int*)d_in[2];
  const int*   batch  = (const int*)d_in[3];
  // d_in[4] = num_nodes (redundant with in_sizes[0])
  const float* node_W = (const float*)d_in[5];
  const float* node_b = (const float*)d_in[6];
  const float* pe_W   = (const float*)d_in[7];
  const float* pe_b   = (const float*)d_in[8];
  const float* conv_W = (const float*)d_in[9];
  const float* conv_b = (const float*)d_in[10];
  const float* bn_g   = (const float*)d_in[11];
  const float* bn_b   = (const float*)d_in[12];
  const float* hW1    = (const float*)d_in[13];
  const float* hb1    = (const float*)d_in[14];
  const float* hW2    = (const float*)d_in[15];
  const float* hb2    = (const float*)d_in[16];

  const int N = in_sizes[0];        // x is [N,1]
  const int E = in_sizes[2] / 2;    // edge_index is [2,E]
  const int NC = N * C_DIM;

  // workspace layout (floats)
  float* ws     = (float*)d_ws;
  float* h      = ws;                       // [N, 128]
  float* tmp    = h + (size_t)NC;           // [N, 128]
  float* dinv   = tmp + (size_t)NC;         // [N]  (degree, then deg^-1/2 in place)
  float* bnsum  = dinv + N;                 // [128]
  float* bnsq   = bnsum + C_DIM;            // [128]
  float* pooled = bnsq + C_DIM;             // [512, 128]
  float* cnt    = pooled + NUM_G * C_DIM;   // [512]

  const int T = 256;
  // encoder
  k_init_h<<<(NC + T - 1) / T, T, 0, stream>>>(x, pe, node_W, node_b, pe_W, pe_b, h, N);
  // degrees -> dinv
  k_zero<<<(N + T - 1) / T, T, 0, stream>>>(dinv, N);
  k_degree<<<(E + T - 1) / T, T, 0, stream>>>(ei, dinv, E);
  k_dinv<<<(N + T - 1) / T, T, 0, stream>>>(dinv, N);

  for (int l = 0; l < 3; ++l) {
    // agg starts as h (residual self-term), then scatter adds messages
    k_copy4<<<(NC / 4 + T - 1) / T, T, 0, stream>>>((const float4*)h, (float4*)tmp, NC / 4);
    k_zero<<<1, 256, 0, stream>>>(bnsum, 2 * C_DIM);  // bnsum + bnsq contiguous
    k_edge<<<((size_t)E * 32 + T - 1) / T, T, 0, stream>>>(ei, dinv, h, tmp, E);
    // tmp = tmp @ W_l + b_l (WMMA, in place) with fused BN stats
    k_gemm_bn<<<(N + 127) / 128, 256, 0, stream>>>(tmp, conv_W + (size_t)l * C_DIM * C_DIM,
                                                   conv_b + l * C_DIM, tmp, bnsum, bnsq, N);
    // h += relu(BN(tmp))
    k_bn_apply<<<(NC + T - 1) / T, T, 0, stream>>>(tmp, h, bnsum, bnsq, bn_g + l * C_DIM,
                                                   bn_b + l * C_DIM, 1.0f / (float)N, NC);
  }

  // pooling
  k_zero<<<(NUM_G * C_DIM + NUM_G + T - 1) / T, T, 0, stream>>>(pooled, NUM_G * C_DIM + NUM_G);
  k_pool<<<((size_t)N * 32 + T - 1) / T, T, 0, stream>>>(h, batch, pooled, cnt, N);

  // head
  k_head<<<NUM_G, C_DIM, 0, stream>>>(pooled, cnt, hW1, hb1, hW2, hb2, (float*)d_out);
}